// Net_51599737094283
// MI455X (gfx1250) — compile-verified
//
#include <hip/hip_runtime.h>
#include <hip/hip_bf16.h>

#define HE_COUNT 20000

typedef __attribute__((ext_vector_type(2))) float v2f;
typedef __attribute__((ext_vector_type(8))) float v8f;

// ---------------- float <-> order-preserving uint (for atomic float max) -----
__device__ __forceinline__ unsigned f2o(float f) {
    unsigned u = __float_as_uint(f);
    return (u & 0x80000000u) ? ~u : (u | 0x80000000u);
}
__device__ __forceinline__ float o2f(unsigned o) {
    return (o & 0x80000000u) ? __uint_as_float(o & 0x7fffffffu)
                             : __uint_as_float(~o);
}
// ordered encoding of -inf : bits(-inf)=0xFF800000 -> ~ = 0x007FFFFF
#define ORD_NEG_INF 0x007FFFFFu

// ---------------- utility fills ---------------------------------------------
__global__ void k_fill_f(float* p, float v, long n) {
    long i = (long)blockIdx.x * blockDim.x + threadIdx.x;
    if (i < n) p[i] = v;
}
__global__ void k_fill_u(unsigned* p, unsigned v, long n) {
    long i = (long)blockIdx.x * blockDim.x + threadIdx.x;
    if (i < n) p[i] = v;
}

// ---------------- WMMA fp32 GEMM: C[M,N] = act(A[M,K] @ B[K,N] + bias) -------
// Block = 8 waves. The K x 64 B-slab is staged into LDS once per block
// (float4 global loads -> ds_store_b128), then every wave reads its B
// fragments from LDS (conflict-free: lanes 0-15 / 16-31 hit disjoint rows).
// Each wave owns one 16-row M tile and computes a 16x64 slab as 4 adjacent
// 16x16 N-tiles sharing the A fragment (4 x v_wmma_f32_16x16x4_f32 / k-step).
// A-frag layout (V_WMMA_F32_16X16X4_F32): lanes 0-15 -> K=0,1 ; 16-31 -> K=2,3.
__global__ __launch_bounds__(256) void k_gemm16x64(
    const float* __restrict__ A, const float* __restrict__ B,
    const float* __restrict__ bias, float* __restrict__ C,
    int M, int K, int Ncols, int relu)
{
    __shared__ float Bs[256 * 64];            // max K = 256 -> 64 KB

    const int tn0 = blockIdx.y * 64;

    // ---- stage B[K, tn0:tn0+64] into LDS (all 256 threads) ----
    for (int idx = threadIdx.x; idx < K * 16; idx += 256) {
        int k = idx >> 4;
        int c = (idx & 15) * 4;
        const float4 v = *(const float4*)(B + (size_t)k * Ncols + tn0 + c);
        *(float4*)&Bs[k * 64 + c] = v;
    }
    __syncthreads();

    const int wave = threadIdx.x >> 5;
    const int lane = threadIdx.x & 31;
    const int half = lane >> 4;               // 0: K=0,1   1: K=2,3
    const int lrow = lane & 15;
    const int tm16 = (blockIdx.x * 8 + wave) * 16;
    if (tm16 >= M) return;                    // after the only barrier

    v8f acc[4] = {};

    const float* Arow = A + (size_t)(tm16 + lrow) * K + half * 2;
    for (int k = 0; k < K; k += 4) {
        __builtin_prefetch(Arow + k + 32, 0, 0);   // global_prefetch_b8
        v2f a;
        a.x = Arow[k + 0];
        a.y = Arow[k + 1];
#pragma unroll
        for (int t = 0; t < 4; ++t) {
            const float* Bp = &Bs[(k + half * 2) * 64 + t * 16 + lrow];
            v2f b;
            b.x = Bp[0];
            b.y = Bp[64];
            acc[t] = __builtin_amdgcn_wmma_f32_16x16x4_f32(
                false, a, false, b, (short)0, acc[t], false, false);
        }
    }

#pragma unroll
    for (int t = 0; t < 4; ++t) {
        int col = tn0 + t * 16 + lrow;
        float bv = bias ? bias[col] : 0.0f;
#pragma unroll
        for (int v = 0; v < 8; ++v) {
            int row = tm16 + half * 8 + v;    // C/D layout: VGPR v -> M = v (+8 hi half)
            float val = acc[t][v] + bv;
            if (relu) val = fmaxf(val, 0.0f);
            C[(size_t)row * Ncols + col] = val;
        }
    }
}

// ---------------- degree count / reciprocal ----------------------------------
__global__ void k_count(const int* __restrict__ idx, float* __restrict__ deg, int n) {
    int i = blockIdx.x * blockDim.x + threadIdx.x;
    if (i < n) atomicAdd(&deg[idx[i]], 1.0f);
}
__global__ void k_recip(float* p, int n) {
    int i = blockIdx.x * blockDim.x + threadIdx.x;
    if (i < n) p[i] = (p[i] > 0.0f) ? (1.0f / p[i]) : 0.0f;
}

// ---------------- generic 4-wide scatter-add ---------------------------------
// dst[d[i]*C + c] += scale[s[i]] * src[s[i]*C + c]
__global__ void k_scatter4(const float* __restrict__ src,
                           const int* __restrict__ sidx,
                           const int* __restrict__ didx,
                           const float* __restrict__ scale,
                           float* __restrict__ dst, int nInc, int C)
{
    long t = (long)blockIdx.x * blockDim.x + threadIdx.x;
    long total = (long)nInc * (C / 4);
    if (t >= total) return;
    int i = (int)(t / (C / 4));
    int c = (int)(t % (C / 4)) * 4;
    int s = sidx[i], d = didx[i];
    float sc = scale ? scale[s] : 1.0f;
    const float* sp = src + (size_t)s * C + c;
    float* dp = dst + (size_t)d * C + c;
#pragma unroll
    for (int q = 0; q < 4; ++q) atomicAdd(dp + q, sc * sp[q]);
}

// ---------------- hypergraph finalize: h = relu(Dinv[n]*h + b[c]) -----------
__global__ void k_hg_final(float* __restrict__ h, const float* __restrict__ Dinv,
                           const float* __restrict__ b, int N, int C)
{
    long t = (long)blockIdx.x * blockDim.x + threadIdx.x;
    if (t >= (long)N * C) return;
    int n = (int)(t / C), c = (int)(t % C);
    h[t] = fmaxf(Dinv[n] * h[t] + b[c], 0.0f);
}

// ---------------- GAT attention scores: asrc = hW@a_s, adst = hW@a_d ---------
// One wave per node: coalesced row reads + cross-lane shuffle reduction.
__global__ void k_attn(const float* __restrict__ hW, const float* __restrict__ as,
                       const float* __restrict__ ad, float* __restrict__ os,
                       float* __restrict__ od, int N, int C)
{
    int wid  = (int)(((long)blockIdx.x * blockDim.x + threadIdx.x) >> 5);
    int lane = threadIdx.x & 31;
    if (wid >= N) return;
    const float* hp = hW + (size_t)wid * C;
    float s = 0.0f, d = 0.0f;
    for (int c = lane; c < C; c += 32) {
        float v = hp[c];
        s += v * as[c];
        d += v * ad[c];
    }
#pragma unroll
    for (int m = 16; m; m >>= 1) {
        s += __shfl_xor(s, m, 32);
        d += __shfl_xor(d, m, 32);
    }
    if (lane == 0) { os[wid] = s; od[wid] = d; }
}

// edge i in [0, E+N): real edges then self-loops
__device__ __forceinline__ void edge_sd(const int* ei, int E, int i, int& s, int& d) {
    if (i < E) { s = ei[i]; d = ei[E + i]; } else { s = i - E; d = i - E; }
}
__device__ __forceinline__ float leaky02(float v) { return v > 0.0f ? v : 0.2f * v; }

__global__ void k_edge_max(const int* __restrict__ ei, int E, int N,
                           const float* __restrict__ asrc, const float* __restrict__ adst,
                           unsigned* __restrict__ Mx)
{
    int i = blockIdx.x * blockDim.x + threadIdx.x;
    if (i >= E + N) return;
    int s, d; edge_sd(ei, E, i, s, d);
    float l = leaky02(asrc[s] + adst[d]);
    atomicMax(&Mx[d], f2o(l));
}

__global__ void k_edge_exp(const int* __restrict__ ei, int E, int N,
                           const float* __restrict__ asrc, const float* __restrict__ adst,
                           const unsigned* __restrict__ Mx, float* __restrict__ e,
                           float* __restrict__ S)
{
    int i = blockIdx.x * blockDim.x + threadIdx.x;
    if (i >= E + N) return;
    int s, d; edge_sd(ei, E, i, s, d);
    float l = leaky02(asrc[s] + adst[d]);
    float m = o2f(Mx[d]);
    if (!isfinite(m)) m = 0.0f;
    float ex = expf(l - m);
    e[i] = ex;
    atomicAdd(&S[d], ex);
}

__global__ void k_edge_agg(const int* __restrict__ ei, int E, int N,
                           const float* __restrict__ e, const float* __restrict__ S,
                           const float* __restrict__ hW, float* __restrict__ acc, int C)
{
    long t = (long)blockIdx.x * blockDim.x + threadIdx.x;
    long total = (long)(E + N) * (C / 4);
    if (t >= total) return;
    int i = (int)(t / (C / 4));
    int c = (int)(t % (C / 4)) * 4;
    int s, d; edge_sd(ei, E, i, s, d);
    float al = e[i] / (S[d] + 1e-16f);
    const float* hp = hW + (size_t)s * C + c;
    float* ap = acc + (size_t)d * C + c;
#pragma unroll
    for (int q = 0; q < 4; ++q) atomicAdd(ap + q, al * hp[q]);
}

__global__ void k_bias_relu(const float* __restrict__ acc, const float* __restrict__ b,
                            float* __restrict__ out, long total, int C)
{
    long t = (long)blockIdx.x * blockDim.x + threadIdx.x;
    if (t >= total) return;
    out[t] = fmaxf(acc[t] + b[t % C], 0.0f);
}

// ---------------- decoder: gather concat(z[src], z[dst]) --------------------
__global__ void k_gather(const float* __restrict__ z, const int* __restrict__ eli,
                         int L, float* __restrict__ eattr)
{
    long t = (long)blockIdx.x * blockDim.x + threadIdx.x;
    long total = (long)L * 32;               // 128 floats per edge / 4
    if (t >= total) return;
    int i = (int)(t / 32);
    int q = (int)(t % 32);
    int which = q >> 4;                      // 0: src half, 1: dst half
    int c4 = (q & 15) * 4;
    int node = (which == 0) ? eli[i] : eli[L + i];
    const float* zp = z + (size_t)node * 64 + c4;
    float* op = eattr + (size_t)i * 128 + which * 64 + c4;
    op[0] = zp[0]; op[1] = zp[1]; op[2] = zp[2]; op[3] = zp[3];
}

// ---------------- decoder head: 64->3, argmax, prob, logits -----------------
__global__ void k_head(const float* __restrict__ h3, const float* __restrict__ Wm4,
                       const float* __restrict__ bm4, float* __restrict__ out, int L)
{
    int i = blockIdx.x * blockDim.x + threadIdx.x;
    if (i >= L) return;
    const float* hp = h3 + (size_t)i * 64;
    float l0 = bm4[0], l1 = bm4[1], l2 = bm4[2];
    for (int c = 0; c < 64; ++c) {
        float v = hp[c];
        l0 += v * Wm4[c * 3 + 0];
        l1 += v * Wm4[c * 3 + 1];
        l2 += v * Wm4[c * 3 + 2];
    }
    int p = 0; float best = l0;
    if (l1 > best) { best = l1; p = 1; }
    if (l2 > best) { best = l2; p = 2; }
    out[i]     = (float)p;                         // out_pred
    out[L + i] = best;                             // out_prob
    float* lg = out + (size_t)2 * L + (size_t)i * 3;
    lg[0] = l0; lg[1] = l1; lg[2] = l2;            // out logits
}

// ---------------- host-side helpers -----------------------------------------
static inline dim3 gsz(long n, int b) { return dim3((unsigned)((n + b - 1) / b)); }

static void launch_gemm(hipStream_t stream, const float* A, const float* B,
                        const float* bias, float* C, int M, int K, int Ncols, int relu)
{
    dim3 grid((unsigned)((M / 16 + 7) / 8), (unsigned)(Ncols / 64));
    k_gemm16x64<<<grid, 256, 0, stream>>>(A, B, bias, C, M, K, Ncols, relu);
}

static void run_gat(hipStream_t stream, const float* h_in, int K, int C,
                    const float* Wt, const float* avs, const float* avd, const float* bias,
                    float* hW, float* acc, float* asrc, float* adst,
                    unsigned* Mx, float* S, float* ebuf,
                    const int* ei, int E, int N, float* out)
{
    launch_gemm(stream, h_in, Wt, nullptr, hW, N, K, C, 0);
    k_attn<<<gsz((long)N * 32, 256), 256, 0, stream>>>(hW, avs, avd, asrc, adst, N, C);
    k_fill_u<<<gsz(N, 256), 256, 0, stream>>>(Mx, ORD_NEG_INF, N);
    k_fill_f<<<gsz(N, 256), 256, 0, stream>>>(S, 0.0f, N);
    k_fill_f<<<gsz((long)N * C, 256), 256, 0, stream>>>(acc, 0.0f, (long)N * C);
    long T = (long)E + N;
    k_edge_max<<<gsz(T, 256), 256, 0, stream>>>(ei, E, N, asrc, adst, Mx);
    k_edge_exp<<<gsz(T, 256), 256, 0, stream>>>(ei, E, N, asrc, adst, Mx, ebuf, S);
    k_edge_agg<<<gsz(T * (C / 4), 256), 256, 0, stream>>>(ei, E, N, ebuf, S, hW, acc, C);
    k_bias_relu<<<gsz((long)N * C, 256), 256, 0, stream>>>(acc, bias, out, (long)N * C, C);
}

extern "C" void kernel_launch(void* const* d_in, const int* in_sizes, int n_in,
                              void* d_out, int out_size, void* d_ws, size_t ws_size,
                              hipStream_t stream)
{
    const float* x   = (const float*)d_in[0];
    const int*   ei  = (const int*)d_in[1];
    const int*   hen = (const int*)d_in[2];
    const int*   hee = (const int*)d_in[3];
    const int*   eli = (const int*)d_in[4];
    const float* W1  = (const float*)d_in[5];
    const float* b1  = (const float*)d_in[6];
    const float* W2  = (const float*)d_in[7];
    const float* a2s = (const float*)d_in[8];
    const float* a2d = (const float*)d_in[9];
    const float* b2  = (const float*)d_in[10];
    const float* W3  = (const float*)d_in[11];
    const float* a3s = (const float*)d_in[12];
    const float* a3d = (const float*)d_in[13];
    const float* b3  = (const float*)d_in[14];
    const float* Wm1 = (const float*)d_in[15];
    const float* bm1 = (const float*)d_in[16];
    const float* Wm2 = (const float*)d_in[17];
    const float* bm2 = (const float*)d_in[18];
    const float* Wm3 = (const float*)d_in[19];
    const float* bm3 = (const float*)d_in[20];
    const float* Wm4 = (const float*)d_in[21];
    const float* bm4 = (const float*)d_in[22];

    const int N  = in_sizes[0] / 128;  // 50000
    const int E  = in_sizes[1] / 2;    // 800000
    const int I  = in_sizes[2];        // 400000 incidences
    const int L  = in_sizes[4] / 2;    // 100000
    const int HE = HE_COUNT;           // 20000

    // -------- workspace layout (floats), heavy reuse, ~158 MB total --------
    float* ws   = (float*)d_ws;
    const size_t BIG = (size_t)N * 256;      // 12.8M floats
    float* h1   = ws;                        // h1 -> h2 -> z live here
    float* hW   = ws + BIG;                  // GEMM outputs / MLP m1,m2,m3
    float* acc  = ws + 2 * BIG;              // m / GAT accumulator / eattr
    float* degD = ws + 3 * BIG;              // [N]
    float* degB = degD + N;                  // [HE]
    float* asrc = degB + HE;                 // [N]
    float* adst = asrc + N;                  // [N]
    unsigned* Mx = (unsigned*)(adst + N);    // [N]
    float* S    = (float*)(Mx + N);          // [N]
    float* ebuf = S + N;                     // [E + N]

    // ================= HypergraphConv =================
    // hX = x @ W1   [N,256]
    launch_gemm(stream, x, W1, nullptr, hW, N, 128, 256, 0);
    // degrees D (node), B (hyperedge) -> reciprocals
    k_fill_f<<<gsz(N, 256), 256, 0, stream>>>(degD, 0.0f, N);
    k_fill_f<<<gsz(HE, 256), 256, 0, stream>>>(degB, 0.0f, HE);
    k_count<<<gsz(I, 256), 256, 0, stream>>>(hen, degD, I);
    k_count<<<gsz(I, 256), 256, 0, stream>>>(hee, degB, I);
    k_recip<<<gsz(N, 256), 256, 0, stream>>>(degD, N);
    k_recip<<<gsz(HE, 256), 256, 0, stream>>>(degB, HE);
    // m[e] = sum_{n in e} hX[n]   (acc holds m: HE x 256)
    k_fill_f<<<gsz((long)HE * 256, 256), 256, 0, stream>>>(acc, 0.0f, (long)HE * 256);
    k_fill_f<<<gsz((long)N * 256, 256), 256, 0, stream>>>(h1, 0.0f, (long)N * 256);
    k_scatter4<<<gsz((long)I * 64, 256), 256, 0, stream>>>(hW, hen, hee, nullptr, acc, I, 256);
    // h1[n] += Binv[e] * m[e]
    k_scatter4<<<gsz((long)I * 64, 256), 256, 0, stream>>>(acc, hee, hen, degB, h1, I, 256);
    // h1 = relu(Dinv * h1 + b1)
    k_hg_final<<<gsz((long)N * 256, 256), 256, 0, stream>>>(h1, degD, b1, N, 256);

    // ================= GAT layer 1 (256 -> 256), h2 overwrites h1 =========
    run_gat(stream, h1, 256, 256, W2, a2s, a2d, b2,
            hW, acc, asrc, adst, Mx, S, ebuf, ei, E, N, h1);

    // ================= GAT layer 2 (256 -> 64), z overwrites h1 ===========
    run_gat(stream, h1, 256, 64, W3, a3s, a3d, b3,
            hW, acc, asrc, adst, Mx, S, ebuf, ei, E, N, h1);

    // ================= Decoder ============================================
    // eattr = concat(z[src], z[dst])  [L,128]  -> acc
    k_gather<<<gsz((long)L * 32, 256), 256, 0, stream>>>(h1, eli, L, acc);
    float* m1 = hW;                       // [L,64]
    float* m2 = hW + (size_t)L * 64;      // [L,64]
    launch_gemm(stream, acc, Wm1, bm1, m1, L, 128, 64, 1);
    launch_gemm(stream, m1,  Wm2, bm2, m2, L,  64, 64, 1);
    launch_gemm(stream, m2,  Wm3, bm3, m1, L,  64, 64, 1);   // m3 -> m1 slot
    // head: 64 -> 3, argmax / prob / logits into d_out
    k_head<<<gsz(L, 256), 256, 0, stream>>>(m1, Wm4, bm4, (float*)d_out, L);
}